// DepthwiseConvLatticeModule_25400436588639
// MI455X (gfx1250) — compile-verified
//
#include <hip/hip_runtime.h>

// Depthwise conv over irregular lattice neighbourhoods (bandwidth-bound gather).
// out[n,c] = sum_k weight[k,c] * values[nbr[n,k],c] + bias[c]
//
// MI455X reasoning: values table (67MB) fits in 192MB L2 -> gathers hit L2;
// HBM floor ~144MB (idx+out+values once) @ 23.3TB/s => memory bound, WMMA
// inapplicable (per-channel diagonal op). Use gfx1250 async global->LDS DMA
// (ASYNCcnt) for parameter/index staging + 128b vector gathers.

#define KF   9
#define CH   64
#define C4   (CH / 4)     // 16 float4 chunks per row
#define VPB  16           // vertices per block (16 lanes each -> 256 threads)
#define TPB  256

#if __has_builtin(__builtin_amdgcn_global_load_async_to_lds_b128)
#define HAVE_ASYNC_LDS 1
#endif

// The async builtin takes typed, address-space-qualified pointers:
//   arg0: int4 AS(1)* (global source), arg1: int4 AS(3)* (LDS dest), imm offset, imm cpol
typedef int v4i __attribute__((ext_vector_type(4)));
typedef __attribute__((address_space(1))) v4i* g_v4i_p;
typedef __attribute__((address_space(3))) v4i* l_v4i_p;

// generic -> AS casts via integer (numerically safe: global generic == AS1 address;
// low 32 bits of a generic LDS pointer are the LDS byte offset on gfx12xx)
#define AS1P(p) ((g_v4i_p)(unsigned long long)(p))
#define AS3P(p) ((l_v4i_p)(unsigned int)(unsigned long long)(p))

__global__ __launch_bounds__(TPB) void dwconv_lattice_kernel(
    const float* __restrict__ values,   // [N, CH]
    const int*   __restrict__ nidx,     // [N, KF]
    const float* __restrict__ weight,   // [KF, CH]
    const float* __restrict__ bias,     // [CH]
    float*       __restrict__ out)      // [N, CH]
{
    __shared__ __align__(16) float s_w[KF * CH];    // 2304 B
    __shared__ __align__(16) float s_b[CH];         //  256 B
    __shared__ __align__(16) int   s_idx[VPB * KF]; //  576 B (144 ints)

    const int t         = threadIdx.x;
    const int blockBase = blockIdx.x * VPB;

#if defined(HAVE_ASYNC_LDS)
    // Stage parameters + this block's neighbour indices with async DMA to LDS.
    // weight: 144 x b128, bias: 16 x b128, indices: 36 x b128 (144 ints).
    if (t < 144) {
        __builtin_amdgcn_global_load_async_to_lds_b128(
            AS1P(weight + t * 4), AS3P(&s_w[t * 4]), 0, 0);
    } else if (t < 160) {
        const int i = t - 144;
        __builtin_amdgcn_global_load_async_to_lds_b128(
            AS1P(bias + i * 4), AS3P(&s_b[i * 4]), 0, 0);
    } else if (t < 196) {
        const int i = t - 160;
        __builtin_amdgcn_global_load_async_to_lds_b128(
            AS1P(nidx + blockBase * KF + i * 4), AS3P(&s_idx[i * 4]), 0, 0);
    }
#if __has_builtin(__builtin_amdgcn_s_wait_asynccnt)
    __builtin_amdgcn_s_wait_asynccnt(0);
#else
    asm volatile("s_wait_asynccnt 0" ::: "memory");
#endif
    __syncthreads();
#else
    for (int i = t; i < KF * CH; i += TPB) s_w[i] = weight[i];
    if (t < CH)       s_b[t]   = bias[t];
    if (t < VPB * KF) s_idx[t] = nidx[blockBase * KF + t];
    __syncthreads();
#endif

    const int g  = t >> 4;          // vertex within block   [0,16)
    const int c4 = t & 15;          // float4 chunk of C     [0,16)
    const int n  = blockBase + g;

    const float4* __restrict__ val4 = (const float4*)values;
    const float4* lw4 = (const float4*)s_w;
    const float4* lb4 = (const float4*)s_b;

    float4 acc = lb4[c4];

    // Read all 9 indices first so the 9 gathers can issue back-to-back.
    int idx[KF];
#pragma unroll
    for (int k = 0; k < KF; ++k) idx[k] = s_idx[g * KF + k];

#pragma unroll
    for (int k = 0; k < KF; ++k) {
        const bool valid = idx[k] >= 0;
        const int  safe  = valid ? idx[k] : 0;
        // 16 lanes x 16B = one contiguous 256B row per vertex (coalesced b128)
        float4 v = val4[(size_t)safe * C4 + c4];
        // branchless boundary mask (cndmask) keeps full MLP; misses hit L2 anyway
        v.x = valid ? v.x : 0.0f;
        v.y = valid ? v.y : 0.0f;
        v.z = valid ? v.z : 0.0f;
        v.w = valid ? v.w : 0.0f;
        const float4 w = lw4[k * C4 + c4];
        acc.x = fmaf(w.x, v.x, acc.x);
        acc.y = fmaf(w.y, v.y, acc.y);
        acc.z = fmaf(w.z, v.z, acc.z);
        acc.w = fmaf(w.w, v.w, acc.w);
    }

    ((float4*)out)[(size_t)n * C4 + c4] = acc;
}

extern "C" void kernel_launch(void* const* d_in, const int* in_sizes, int n_in,
                              void* d_out, int out_size, void* d_ws, size_t ws_size,
                              hipStream_t stream) {
    const float* values = (const float*)d_in[0];   // [N, 64]
    const int*   nidx   = (const int*)d_in[1];     // [N, 9]
    const float* weight = (const float*)d_in[2];   // [9, 64]
    const float* bias   = (const float*)d_in[3];   // [64]
    float*       out    = (float*)d_out;

    const int N = in_sizes[0] / CH;                // 262144
    dim3 grid(N / VPB), block(TPB);
    dwconv_lattice_kernel<<<grid, block, 0, stream>>>(values, nidx, weight, bias, out);
}